// StackAugmentedRNN_67688684585242
// MI455X (gfx1250) — compile-verified
//
#include <hip/hip_runtime.h>
#include <math.h>

// ---- problem dims (from reference) ----
#define V     128
#define H     2048
#define WW    256          // stack width
#define DEPTH 1024
#define H2    (2 * H)      // 4096
#define RNNL  (H + WW)     // 2304
#define GATES (4 * H)      // 8192

// ---- workspace layout (float offsets into d_ws; needs ~96KB) ----
#define WS_CTRL    0       // 3 floats: a_push, a_pop, a_noop (post-softmax)
#define WS_STACKIN 8       // 256 floats: tanh(W_sin @ h2s + b)
#define WS_RNNIN   512     // 2304 floats: [emb_row(2048), stack_top(256)], 16B aligned
#define WS_G       4096    // 16384 floats: gates fwd(8192) then bwd(8192)
#define WS_OUT     20480   // 4096 floats: [h_f, h_b]

typedef __attribute__((ext_vector_type(2))) float v2f;
typedef __attribute__((ext_vector_type(8))) float v8f;

__device__ __forceinline__ float wave_reduce(float v) {
    #pragma unroll
    for (int m = 16; m > 0; m >>= 1) v += __shfl_xor(v, m, 32);
    return v;
}
__device__ __forceinline__ float sigmoidf(float x) { return 1.0f / (1.0f + expf(-x)); }

// ---- K1a: ctrl logits (3 x 4096 GEMV) + softmax, one block, 3 waves ----
__global__ void ctrl_kernel(const float* __restrict__ Wc, const float* __restrict__ bc,
                            const float* __restrict__ h2s, float* __restrict__ ws) {
    __shared__ float slog[3];
    const int wave = threadIdx.x >> 5, lane = threadIdx.x & 31;
    const float4* wr = (const float4*)(Wc + (size_t)wave * H2);
    const float4* xv = (const float4*)h2s;
    float acc = 0.0f;
    for (int i = lane; i < H2 / 4; i += 32) {
        float4 a = wr[i], b = xv[i];
        acc = fmaf(a.x, b.x, fmaf(a.y, b.y, fmaf(a.z, b.z, fmaf(a.w, b.w, acc))));
    }
    acc = wave_reduce(acc);
    if (lane == 0) slog[wave] = acc + bc[wave];
    __syncthreads();
    if (threadIdx.x == 0) {
        float l0 = slog[0], l1 = slog[1], l2 = slog[2];
        float m = fmaxf(l0, fmaxf(l1, l2));
        float e0 = expf(l0 - m), e1 = expf(l1 - m), e2 = expf(l2 - m);
        float s = e0 + e1 + e2;
        ws[WS_CTRL + 0] = e0 / s;   // a_push
        ws[WS_CTRL + 1] = e1 / s;   // a_pop
        ws[WS_CTRL + 2] = e2 / s;   // a_noop
    }
}

// ---- K1b: stack_in = tanh(W_sin @ h2s + b_sin), one wave per row ----
__global__ void stackin_kernel(const float* __restrict__ Ws, const float* __restrict__ bs,
                               const float* __restrict__ h2s, float* __restrict__ ws) {
    const int wave = threadIdx.x >> 5, lane = threadIdx.x & 31;
    const int row = blockIdx.x * 8 + wave;
    const float4* wr = (const float4*)(Ws + (size_t)row * H2);
    const float4* xv = (const float4*)h2s;
    float acc = 0.0f;
    for (int i = lane; i < H2 / 4; i += 32) {
        float4 a = wr[i], b = xv[i];
        acc = fmaf(a.x, b.x, fmaf(a.y, b.y, fmaf(a.z, b.z, fmaf(a.w, b.w, acc))));
    }
    acc = wave_reduce(acc);
    if (lane == 0) ws[WS_STACKIN + row] = tanhf(acc + bs[row]);
}

// ---- K2: stack blend + write new_stack + build rnn_in ----
__global__ void stack_update_kernel(const float* __restrict__ stack,
                                    const float* __restrict__ emb,
                                    const int* __restrict__ inp,
                                    float* __restrict__ ws,
                                    float* __restrict__ out_stack) {
    const int idx = blockIdx.x * blockDim.x + threadIdx.x;
    const float a_push = ws[WS_CTRL + 0];
    const float a_pop  = ws[WS_CTRL + 1];
    const float a_noop = ws[WS_CTRL + 2];
    if (idx < DEPTH * WW) {
        const int d = idx / WW, w = idx % WW;
        const float up   = (d == 0) ? ws[WS_STACKIN + w] : stack[idx - WW];
        const float down = (d < DEPTH - 1) ? stack[idx + WW] : 0.0f;
        const float v = a_noop * stack[idx] + a_push * up + a_pop * down;
        out_stack[idx] = v;
        if (d == 0) ws[WS_RNNIN + H + w] = v;   // stack_top -> rnn_in tail
    } else {
        const int k = idx - DEPTH * WW;
        if (k < H) ws[WS_RNNIN + k] = emb[(size_t)inp[0] * H + k];  // x = emb[inp]
    }
}

// ---- K3: gate GEMVs (bandwidth-critical, 285 MB of weights streamed) ----
// One wave per output row; coalesced b128 loads + prefetch.
__global__ void gates_kernel(const float* __restrict__ wih_f, const float* __restrict__ whh_f,
                             const float* __restrict__ bih_f, const float* __restrict__ bhh_f,
                             const float* __restrict__ wih_b, const float* __restrict__ whh_b,
                             const float* __restrict__ bih_b, const float* __restrict__ bhh_b,
                             const float* __restrict__ hidden,
                             const float* __restrict__ rnn_in,
                             float* __restrict__ g_out) {
    const int wave = threadIdx.x >> 5, lane = threadIdx.x & 31;
    const int row = blockIdx.x * 8 + wave;           // 0..16383
    const int dir = row >> 13;                       // 0 = fwd, 1 = bwd
    const int j   = row & 8191;
    const float* wih = dir ? wih_b : wih_f;
    const float* whh = dir ? whh_b : whh_f;
    const float* bih = dir ? bih_b : bih_f;
    const float* bhh = dir ? bhh_b : bhh_f;

    float acc = 0.0f;
    {   // W_ih[j] . rnn_in  (2304)
        const float4* wr = (const float4*)(wih + (size_t)j * RNNL);
        const float4* xv = (const float4*)rnn_in;
        for (int i = lane; i < RNNL / 4; i += 32) {
            __builtin_prefetch(&wr[i + 32], 0, 1);   // gfx1250 global_prefetch
            float4 a = wr[i], b = xv[i];
            acc = fmaf(a.x, b.x, fmaf(a.y, b.y, fmaf(a.z, b.z, fmaf(a.w, b.w, acc))));
        }
    }
    {   // W_hh[j] . h_dir  (2048)
        const float4* wr = (const float4*)(whh + (size_t)j * H);
        const float4* xv = (const float4*)(hidden + (size_t)dir * H);
        for (int i = lane; i < H / 4; i += 32) {
            __builtin_prefetch(&wr[i + 32], 0, 1);
            float4 a = wr[i], b = xv[i];
            acc = fmaf(a.x, b.x, fmaf(a.y, b.y, fmaf(a.z, b.z, fmaf(a.w, b.w, acc))));
        }
    }
    acc = wave_reduce(acc);
    if (lane == 0) g_out[row] = acc + bih[j] + bhh[j];
}

// ---- K4: LSTM pointwise activations -> new_hidden/new_cell + out vec ----
__global__ void lstm_act_kernel(const float* __restrict__ g, const float* __restrict__ cell,
                                float* __restrict__ out, float* __restrict__ ws_out) {
    const int idx = blockIdx.x * blockDim.x + threadIdx.x;
    if (idx >= H2) return;
    const int dir = idx >> 11, j = idx & (H - 1);
    const float* gd = g + (size_t)dir * GATES;
    const float gi = gd[j], gf = gd[H + j], gg = gd[2 * H + j], go = gd[3 * H + j];
    const float cold = cell[(size_t)dir * H + j];
    const float c2 = sigmoidf(gf) * cold + sigmoidf(gi) * tanhf(gg);
    const float h2 = sigmoidf(go) * tanhf(c2);
    out[V + dir * H + j] = h2;              // new_hidden
    out[V + H2 + dir * H + j] = c2;         // new_cell
    ws_out[dir * H + j] = h2;               // [h_f, h_b] for decoder
}

// ---- K5: decoder logits via V_WMMA_F32_16X16X4_F32 (exact fp32) ----
// One wave per 16-row tile; B = out-vector replicated across all 16 columns;
// result column N=0 extracted from lanes 0 (M=0..7) and 16 (M=8..15).
__global__ void decoder_wmma_kernel(const float* __restrict__ Wd, const float* __restrict__ bd,
                                    const float* __restrict__ ws_out, float* __restrict__ logits) {
    __shared__ float sx[H2];
    const int lane = threadIdx.x & 31;
    for (int i = threadIdx.x; i < H2; i += blockDim.x) sx[i] = ws_out[i];
    __syncthreads();

    const int m0 = blockIdx.x * 16;
    const int row = m0 + (lane & 15);
    const int klane = (lane >> 4) << 1;              // lanes 0-15 -> K0/K1, 16-31 -> K2/K3
    const float* wrow = Wd + (size_t)row * H2;

    v8f c = {};
    for (int k = 0; k < H2; k += 4) {
        const int kk = k + klane;
        v2f a; a.x = wrow[kk]; a.y = wrow[kk + 1];   // A 16x4: VGPR0=K0|K2, VGPR1=K1|K3
        v2f b; b.x = sx[kk];   b.y = sx[kk + 1];     // B 4x16: row k replicated over N
        c = __builtin_amdgcn_wmma_f32_16x16x4_f32(false, a, false, b,
                                                  (short)0, c, false, false);
    }
    if ((lane & 15) == 0) {                          // lane 0 -> M=0..7, lane 16 -> M=8..15
        const int mb = m0 + ((lane >> 4) << 3);
        #pragma unroll
        for (int r = 0; r < 8; ++r) logits[mb + r] = c[r] + bd[mb + r];
    }
}

extern "C" void kernel_launch(void* const* d_in, const int* in_sizes, int n_in,
                              void* d_out, int out_size, void* d_ws, size_t ws_size,
                              hipStream_t stream) {
    const int*   inp    = (const int*)  d_in[0];
    const float* hidden = (const float*)d_in[1];   // (2,1,H) contiguous == h2s
    const float* cell   = (const float*)d_in[2];
    const float* stack  = (const float*)d_in[3];
    const float* emb    = (const float*)d_in[4];
    const float* W_ctrl = (const float*)d_in[5];
    const float* b_ctrl = (const float*)d_in[6];
    const float* W_sin  = (const float*)d_in[7];
    const float* b_sin  = (const float*)d_in[8];
    const float* w_ih_f = (const float*)d_in[9];
    const float* w_hh_f = (const float*)d_in[10];
    const float* b_ih_f = (const float*)d_in[11];
    const float* b_hh_f = (const float*)d_in[12];
    const float* w_ih_b = (const float*)d_in[13];
    const float* w_hh_b = (const float*)d_in[14];
    const float* b_ih_b = (const float*)d_in[15];
    const float* b_hh_b = (const float*)d_in[16];
    const float* W_dec  = (const float*)d_in[17];
    const float* b_dec  = (const float*)d_in[18];

    float* out = (float*)d_out;              // [logits 128][hidden 4096][cell 4096][stack 262144]
    float* ws  = (float*)d_ws;
    float* out_stack = out + V + H2 + H2;    // offset 8320

    ctrl_kernel<<<1, 96, 0, stream>>>(W_ctrl, b_ctrl, hidden, ws);
    stackin_kernel<<<32, 256, 0, stream>>>(W_sin, b_sin, hidden, ws);
    stack_update_kernel<<<(DEPTH * WW + H + 255) / 256, 256, 0, stream>>>(
        stack, emb, inp, ws, out_stack);
    gates_kernel<<<2048, 256, 0, stream>>>(
        w_ih_f, w_hh_f, b_ih_f, b_hh_f,
        w_ih_b, w_hh_b, b_ih_b, b_hh_b,
        hidden, ws + WS_RNNIN, ws + WS_G);
    lstm_act_kernel<<<(H2 + 255) / 256, 256, 0, stream>>>(ws + WS_G, cell, out, ws + WS_OUT);
    decoder_wmma_kernel<<<8, 32, 0, stream>>>(W_dec, b_dec, ws + WS_OUT, out);
}